// Attention_58437325029953
// MI455X (gfx1250) — compile-verified
//
#include <hip/hip_runtime.h>

// MI455X / gfx1250, wave32. Matrix ops via V_WMMA_F32_16X16X4_F32 (fp32 path,
// matches reference precision). Hot loop (50 projected-gradient iterations)
// runs entirely out of LDS: Q1 repacked as K-pairs (1 ds_load_b64 per B-frag,
// bank-conflict-free) + x tile; x_old carried in registers.

typedef __attribute__((ext_vector_type(2))) float v2f;
typedef __attribute__((ext_vector_type(8))) float v8f;

static __device__ __forceinline__ v8f wmma_f32(v2f a, v2f b, v8f c) {
  // 8 args: (neg_a, A, neg_b, B, c_mod, C, reuse_a, reuse_b)
  return __builtin_amdgcn_wmma_f32_16x16x4_f32(false, a, false, b, (short)0, c,
                                               false, false);
}

#define B_  4
#define N_  2048
#define M_  256
#define D_  256
#define XS  260      // x-tile LDS row stride: 260 % 64 == 4 -> b64 conflict-free
#define QPS 544      // packed-Q1 LDS stride per K-pair: 544 % 64 == 32 ->
                     // lane halves (pairIdx, pairIdx+1) hit disjoint bank sets
#define NT  32       // n-rows per workgroup in solver
#define NITERS 50

// ---------------------------------------------------------------------------
// Kernel 1: Q1[b] = 2 * Vs Vs^T, Vs = V/m  ->  (2/m^2) * V V^T.  One wave per
// 16x16 output tile.
// ---------------------------------------------------------------------------
__global__ __launch_bounds__(32) void k_q1(const float* __restrict__ V,
                                           float* __restrict__ Q1) {
  const int b  = blockIdx.z;
  const int i0 = blockIdx.y * 16;
  const int j0 = blockIdx.x * 16;
  const int lane = threadIdx.x & 31;
  const int half = lane >> 4;     // selects K pair {0,1} vs {2,3}
  const int lr   = lane & 15;     // M for A-frag, N for B-frag
  const float* Vb = V + (size_t)b * M_ * D_;
  v8f acc = {};
  for (int k0 = 0; k0 < D_; k0 += 4) {
    v2f a, bb;
    const float* ar = Vb + (i0 + lr) * D_ + k0 + 2 * half;
    a.x = ar[0]; a.y = ar[1];
    const float* br = Vb + (j0 + lr) * D_ + k0 + 2 * half;   // B[k][j]=V[j][k]
    bb.x = br[0]; bb.y = br[1];
    acc = wmma_f32(a, bb, acc);
  }
  const float sc = 2.0f / ((float)M_ * (float)M_);
  float* out = Q1 + (size_t)b * M_ * M_;
#pragma unroll
  for (int j = 0; j < 8; ++j) {
    int row = i0 + j + 8 * half;
    out[row * M_ + j0 + lr] = acc[j] * sc;
  }
}

// ---------------------------------------------------------------------------
// Kernel 2: L[b] = max_k sum_j |Q1[b,k,j]| + 1e-10
// ---------------------------------------------------------------------------
__global__ __launch_bounds__(256) void k_lmax(const float* __restrict__ Q1,
                                              float* __restrict__ L) {
  __shared__ float red[256];
  const int b = blockIdx.x;
  const int t = threadIdx.x;
  const float* row = Q1 + ((size_t)b * M_ + t) * M_;
  float s = 0.0f;
  for (int j = 0; j < M_; ++j) s += fabsf(row[j]);
  red[t] = s;
  __syncthreads();
  for (int off = 128; off > 0; off >>= 1) {
    if (t < off) red[t] = fmaxf(red[t], red[t + off]);
    __syncthreads();
  }
  if (t == 0) L[b] = red[0] + 1e-10f;
}

// ---------------------------------------------------------------------------
// Kernel 3: P[b,n,m] = (-2/m) * sum_d V[m,d] Q[n,d] + lambda/m.  V[b] staged
// in LDS (padded stride) so B-fragments (transposed reads) are a single
// conflict-free ds_load_b64.  Block: 16 n-rows x 256 m-cols; 8 waves x 2 tiles.
// ---------------------------------------------------------------------------
__global__ __launch_bounds__(256) void k_p(const float* __restrict__ Q,
                                           const float* __restrict__ V,
                                           float* __restrict__ P) {
  extern __shared__ float vlds[];               // M_ * XS floats
  const int b  = blockIdx.y;
  const int n0 = blockIdx.x * 16;
  const float* Vb = V + (size_t)b * M_ * D_;
  const float* Qb = Q + (size_t)b * N_ * D_;
  for (int i = threadIdx.x; i < M_ * D_; i += 256) {
    int r = i >> 8, c = i & 255;
    vlds[r * XS + c] = Vb[i];
  }
  __syncthreads();
  const int wave = threadIdx.x >> 5;
  const int lane = threadIdx.x & 31;
  const int half = lane >> 4, lr = lane & 15;
  v8f acc[2] = {};
  for (int k0 = 0; k0 < D_; k0 += 4) {
    v2f a;
    const float* ar = Qb + (n0 + lr) * D_ + k0 + 2 * half;
    a.x = ar[0]; a.y = ar[1];
#pragma unroll
    for (int t = 0; t < 2; ++t) {
      const int j0 = (wave * 2 + t) * 16;       // B[k][j] = V[j][k]
      v2f bb = *(const v2f*)(vlds + (j0 + lr) * XS + k0 + 2 * half);
      acc[t] = wmma_f32(a, bb, acc[t]);
    }
  }
  const float sc  = -2.0f / (float)M_;
  const float add = 0.1f / (float)M_;           // LAMBDA / m
  float* Pb = P + (size_t)b * N_ * M_;
#pragma unroll
  for (int t = 0; t < 2; ++t) {
    const int j0 = (wave * 2 + t) * 16;
#pragma unroll
    for (int j = 0; j < 8; ++j) {
      int row = n0 + j + 8 * half;
      Pb[(size_t)row * M_ + j0 + lr] = acc[t][j] * sc + add;
    }
  }
}

// ---------------------------------------------------------------------------
// Kernel 4: 50 projected-gradient iterations + normalize + out = coeffs * Vs.
// LDS: Q1[b] packed as K-pairs (128 x QPS) + x tile (NT x XS) + row-sum
// scratch.  Wave w owns row-block rb = w&1 and col-tiles cb0..cb0+3.
// ---------------------------------------------------------------------------
__global__ __launch_bounds__(256) void k_solve(const float* __restrict__ Q1g,
                                               const float* __restrict__ Lg,
                                               const float* __restrict__ Pg,
                                               const float* __restrict__ Vg,
                                               float* __restrict__ out) {
  extern __shared__ float smem[];
  float* q1   = smem;                       // 128 * QPS floats (packed pairs)
  float* x    = smem + 128 * QPS;           // NT * XS floats
  float* rsum = x + NT * XS;                // 256 floats
  const int b  = blockIdx.y;
  const int n0 = blockIdx.x * NT;

  // Load Q1[b] and repack as K-pairs: q1[(k/2)*QPS + col*2 + (k&1)].
  const float* Q1b = Q1g + (size_t)b * M_ * M_;
  for (int i = threadIdx.x; i < M_ * M_; i += 256) {
    int k = i >> 8, c = i & 255;
    q1[(k >> 1) * QPS + c * 2 + (k & 1)] = Q1b[i];
  }
  for (int i = threadIdx.x; i < NT * XS; i += 256) x[i] = 0.0f;

  const float invL = 1.0f / Lg[b];
  const int wave = threadIdx.x >> 5;
  const int lane = threadIdx.x & 31;
  const int half = lane >> 4, lr = lane & 15;
  const int rb  = wave & 1;
  const int cb0 = (wave >> 1) * 4;

  // P tile for this wave's output elements, held in registers.
  const float* Pb = Pg + ((size_t)b * N_ + n0) * M_;
  float preg[4][8];
  float xreg[4][8];                         // this wave's x values (init 0)
#pragma unroll
  for (int t = 0; t < 4; ++t)
#pragma unroll
    for (int j = 0; j < 8; ++j) {
      int row = rb * 16 + j + 8 * half;
      int col = (cb0 + t) * 16 + lr;
      preg[t][j] = Pb[(size_t)row * M_ + col];
      xreg[t][j] = 0.0f;
    }

  const float* xrow = x + (rb * 16 + lr) * XS + 2 * half;    // A-frag base
  const float* qbase = q1 + half * QPS + lr * 2;             // B-frag base
  __syncthreads();

  for (int it = 0; it < NITERS; ++it) {
    v8f acc[4] = {};
    for (int k0 = 0; k0 < M_; k0 += 4) {
      v2f a = *(const v2f*)(xrow + k0);                      // ds_load_b64
#pragma unroll
      for (int t = 0; t < 4; ++t) {                          // ds_load_b64
        v2f bb = *(const v2f*)(qbase + (k0 >> 1) * QPS + (cb0 + t) * 32);
        acc[t] = wmma_f32(a, bb, acc[t]);
      }
    }
    __syncthreads();                         // all reads of x done
#pragma unroll
    for (int t = 0; t < 4; ++t)
#pragma unroll
      for (int j = 0; j < 8; ++j) {
        int row = rb * 16 + j + 8 * half;
        int col = (cb0 + t) * 16 + lr;
        float xn = xreg[t][j] - (acc[t][j] + preg[t][j]) * invL;
        xn = fminf(fmaxf(xn, 0.0f), 1.0f);
        xreg[t][j] = xn;
        x[row * XS + col] = xn;
      }
    __syncthreads();                         // x update visible to all
  }

  // Row sums (x >= 0 so sum == sum|x|): 8 partials per row, then invert.
  {
    int row = threadIdx.x >> 3, seg = threadIdx.x & 7;
    float s = 0.0f;
    const float* xr = x + row * XS + seg * 32;
    for (int j = 0; j < 32; ++j) s += xr[j];
    rsum[row * 8 + seg] = s;
  }
  __syncthreads();
  if (threadIdx.x < NT) {
    float s = 0.0f;
#pragma unroll
    for (int j = 0; j < 8; ++j) s += rsum[threadIdx.x * 8 + j];
    rsum[threadIdx.x] = 1.0f / (s + 1e-10f); // lockstep within one wave
  }
  __syncthreads();

  // out[n, d] = rsum[n] * (1/m) * sum_k x[n,k] * V[b,k,d]  (V row-major ->
  // coalesced B-fragment loads straight from global).
  const float* Vb = Vg + (size_t)b * M_ * D_;
  v8f oac[4] = {};
  for (int k0 = 0; k0 < M_; k0 += 4) {
    v2f a = *(const v2f*)(xrow + k0);
#pragma unroll
    for (int t = 0; t < 4; ++t) {
      v2f bb;
      const float* br = Vb + (size_t)(k0 + 2 * half) * D_ + (cb0 + t) * 16 + lr;
      bb.x = br[0]; bb.y = br[D_];
      oac[t] = wmma_f32(a, bb, oac[t]);
    }
  }
  float* outb = out + ((size_t)b * N_ + n0) * D_;
  const float im = 1.0f / (float)M_;
#pragma unroll
  for (int t = 0; t < 4; ++t)
#pragma unroll
    for (int j = 0; j < 8; ++j) {
      int row = rb * 16 + j + 8 * half;
      int col = (cb0 + t) * 16 + lr;
      outb[(size_t)row * D_ + col] = oac[t][j] * rsum[row] * im;
    }
}

// ---------------------------------------------------------------------------
extern "C" void kernel_launch(void* const* d_in, const int* in_sizes, int n_in,
                              void* d_out, int out_size, void* d_ws, size_t ws_size,
                              hipStream_t stream) {
  const float* Q = (const float*)d_in[0];   // [4,2048,256]
  const float* V = (const float*)d_in[1];   // [4,256,256]
  float* out = (float*)d_out;               // [4,2048,256]

  // Workspace layout (floats): Q1 | L(pad 64) | P   ~= 9.5 MB total
  float* q1 = (float*)d_ws;                 // 4*256*256   = 262144
  float* L  = q1 + (size_t)B_ * M_ * M_;    // 4 (padded to 64)
  float* P  = L + 64;                       // 4*2048*256  = 2097152

  k_q1<<<dim3(M_ / 16, M_ / 16, B_), 32, 0, stream>>>(V, q1);
  k_lmax<<<B_, 256, 0, stream>>>(q1, L);
  k_p<<<dim3(N_ / 16, B_), 256, (size_t)M_ * XS * sizeof(float), stream>>>(Q, V, P);
  const size_t solve_lds =
      (size_t)(128 * QPS + NT * XS + 256) * sizeof(float);   // ~305 KB
  k_solve<<<dim3(N_ / NT, B_), 256, solve_lds, stream>>>(q1, L, P, V, out);
}